// HamiltonianFlow_23957327577411
// MI455X (gfx1250) — compile-verified
//
#include <hip/hip_runtime.h>
#include <stdint.h>

// ---------------------------------------------------------------------------
// Hamiltonian flow: H = sum(0.5 p^2 + 0.5 q^2 + 0.25 q^4)  (separable!)
//   dq/dt = p
//   dp/dt = -(q + q^3)
// 255 RK4 steps, dt = 10/255. Every (q,p) pair is independent -> pure VALU.
// Strategy: packed-FP32 (v_pk_fma_f32) register-resident integration,
// 4 pairs per thread as 2 independent packed chains, async-to-LDS staged load.
// ---------------------------------------------------------------------------

typedef __attribute__((ext_vector_type(2))) float f32x2;

__device__ __forceinline__ f32x2 pk_fma(f32x2 a, f32x2 b, f32x2 c) {
    f32x2 d;
    asm("v_pk_fma_f32 %0, %1, %2, %3" : "=v"(d) : "v"(a), "v"(b), "v"(c));
    return d;
}
__device__ __forceinline__ f32x2 pk_mul(f32x2 a, f32x2 b) {
    f32x2 d;
    asm("v_pk_mul_f32 %0, %1, %2" : "=v"(d) : "v"(a), "v"(b));
    return d;
}
__device__ __forceinline__ f32x2 pk_add(f32x2 a, f32x2 b) {
    f32x2 d;
    asm("v_pk_add_f32 %0, %1, %2" : "=v"(d) : "v"(a), "v"(b));
    return d;
}

// g(u) = u + u^3 = fma(u*u, u, u)
__device__ __forceinline__ f32x2 g_of(f32x2 u) {
    return pk_fma(pk_mul(u, u), u, u);
}

__global__ __launch_bounds__(256)
void HamiltonianFlow_rk4_kernel(const float4* __restrict__ x0,
                                float4* __restrict__ xT,
                                int n4half) {
    __shared__ float4 tile[512];

    const int lt = threadIdx.x;
    const int t  = blockIdx.x * 256 + lt;
    if (t >= n4half) return;

    // ---- Stage the one-time input load through LDS via the CDNA5 async path.
    // Generic LDS pointer: low 32 bits are the LDS byte offset (ISA 10.2).
    unsigned ldsA = (unsigned)(uintptr_t)(&tile[lt]);
    unsigned ldsB = (unsigned)(uintptr_t)(&tile[lt + 256]);
    unsigned long long gA = (unsigned long long)(uintptr_t)(x0 + t);
    unsigned long long gB = (unsigned long long)(uintptr_t)(x0 + t + n4half);
    asm volatile("global_load_async_to_lds_b128 %0, %1, off"
                 :: "v"(ldsA), "v"(gA) : "memory");
    asm volatile("global_load_async_to_lds_b128 %0, %1, off"
                 :: "v"(ldsB), "v"(gB) : "memory");
    asm volatile("s_wait_asynccnt 0x0" ::: "memory");
    // Each lane reads back only bytes it wrote itself -> no barrier needed.
    float4 va = tile[lt];
    float4 vb = tile[lt + 256];

    // Deinterleave (q,p,q,p) into packed q / packed p per chain.
    f32x2 q[2], p[2];
    q[0] = (f32x2){va.x, va.z};  p[0] = (f32x2){va.y, va.w};
    q[1] = (f32x2){vb.x, vb.z};  p[1] = (f32x2){vb.y, vb.w};

    // RK4 coefficients (float32, matching jnp dt = 10/255 in f32).
    const float dt  = 10.0f / 255.0f;
    const float ha  = 0.5f * dt;
    const float dt6 = dt / 6.0f;
    const f32x2 vdt   = (f32x2){ dt,   dt  };
    const f32x2 vmdt  = (f32x2){-dt,  -dt  };
    const f32x2 vha   = (f32x2){ ha,   ha  };
    const f32x2 vmha  = (f32x2){-ha,  -ha  };
    const f32x2 vdt6  = (f32x2){ dt6,  dt6 };
    const f32x2 vmdt6 = (f32x2){-dt6, -dt6 };
    const f32x2 v2    = (f32x2){ 2.0f, 2.0f};

    // 255 sequential RK4 steps; two independent packed chains interleaved
    // per iteration for ILP. All state stays in VGPRs.
    #pragma unroll 1
    for (int s = 0; s < 255; ++s) {
        #pragma unroll
        for (int c = 0; c < 2; ++c) {
            // k1: (p, -g(q))
            f32x2 g1 = g_of(q[c]);
            f32x2 q2 = pk_fma(vha,  p[c], q[c]);   // q + (dt/2) k1q
            f32x2 p2 = pk_fma(vmha, g1,   p[c]);   // p - (dt/2) g1
            // k2
            f32x2 g2 = g_of(q2);
            f32x2 q3 = pk_fma(vha,  p2, q[c]);
            f32x2 p3 = pk_fma(vmha, g2, p[c]);
            // k3
            f32x2 g3 = g_of(q3);
            f32x2 q4 = pk_fma(vdt,  p3, q[c]);
            f32x2 p4 = pk_fma(vmdt, g3, p[c]);
            // k4
            f32x2 g4 = g_of(q4);
            // sums: k1 + 2k2 + 2k3 + k4 (q-part uses p's, p-part uses g's)
            f32x2 sq = pk_fma(v2, p2, p[c]);
            sq       = pk_fma(v2, p3, sq);
            sq       = pk_add(sq, p4);
            f32x2 sp = pk_fma(v2, g2, g1);
            sp       = pk_fma(v2, g3, sp);
            sp       = pk_add(sp, g4);
            q[c] = pk_fma(vdt6,  sq, q[c]);
            p[c] = pk_fma(vmdt6, sp, p[c]);
        }
    }

    // Reinterleave and store.
    float4 oa, ob;
    oa.x = q[0].x; oa.y = p[0].x; oa.z = q[0].y; oa.w = p[0].y;
    ob.x = q[1].x; ob.y = p[1].x; ob.z = q[1].y; ob.w = p[1].y;
    xT[t]          = oa;
    xT[t + n4half] = ob;
}

extern "C" void kernel_launch(void* const* d_in, const int* in_sizes, int n_in,
                              void* d_out, int out_size, void* d_ws, size_t ws_size,
                              hipStream_t stream) {
    const float4* x0 = (const float4*)d_in[0];
    float4* xT = (float4*)d_out;

    // in_sizes[0] = 4096*64*1*2 = 524288 floats -> 131072 float4 (2 pairs each)
    const int nfloats = in_sizes[0];
    const int n4      = nfloats / 4;   // float4 groups
    const int n4half  = n4 / 2;        // each thread handles 2 float4 groups

    const int threads = 256;           // 8 wave32 per block
    const int blocks  = (n4half + threads - 1) / threads;

    hipLaunchKernelGGL(HamiltonianFlow_rk4_kernel,
                       dim3(blocks), dim3(threads), 0, stream,
                       x0, xT, n4half);
}